// Tensor_product_custom_67190468379308
// MI455X (gfx1250) — compile-verified
//
#include <hip/hip_runtime.h>
#include <hip/hip_bf16.h>
#include <math.h>

// ---------------------------------------------------------------------------
// Tensor product (e3nn-style):  out[e,r,c] = sum_{i,j} CG[r,i,j] x[e,i,c] y[e,j]
// Restructured:  W[e] (99x16) = CG . y[e]   (cheap),  out[e] = W[e] @ x[e] (GEMM)
// GEMM runs on V_WMMA_F32_16X16X4_F32 (exact f32 semantics, XDL rate).
// ---------------------------------------------------------------------------

#define N_EDGE_DIM   16      // sum (2l+1), l=0..3
#define N_ROWS       99      // DIM_OUT
#define N_ROWS_PAD   112     // 7 tiles of 16
#define N_CH         64
#define N_PATHS      23
#define W_PITCH      18      // LDS row pitch (floats): conflict-free, 8B aligned

typedef __attribute__((ext_vector_type(2))) float v2f;
typedef __attribute__((ext_vector_type(8))) float v8f;

// Path tables: (l1,l2,l3) with |l1-l2|<=l3<=min(3,l1+l2), l1+l2+l3 even,
// enumerated in the reference's order; PROFF = running row offset (2*l3+1 rows).
__device__ __constant__ int PL1[N_PATHS]  = {0,0,0,0, 1,1,1,1,1,1, 2,2,2,2,2,2,2, 3,3,3,3,3,3};
__device__ __constant__ int PL2[N_PATHS]  = {0,1,2,3, 0,1,1,2,2,3, 0,1,1,2,2,3,3, 0,1,2,2,3,3};
__device__ __constant__ int PL3[N_PATHS]  = {0,1,2,3, 1,0,2,1,3,2, 2,1,3,0,2,1,3, 3,2,1,3,0,2};
__device__ __constant__ int PROFF[N_PATHS]= {0,1,4,9, 16,19,20,25,28,35, 40,45,48,55,56,61,64, 71,78,83,86,93,94};
__device__ __constant__ int SPHOFF[4] = {0,1,4,9};

// ------------------------- CG table construction ---------------------------

static __device__ double factd(int n) {
    double r = 1.0;
    for (int i = 2; i <= n; ++i) r *= (double)i;
    return r;
}

static __device__ double cg_coeff(int j1,int m1,int j2,int m2,int j3,int m3) {
    if (m3 != m1 + m2) return 0.0;
    int vmin = max(0, max(j2 - j3 - m1, j1 - j3 + m2));
    int vmax = min(j1 + j2 - j3, min(j1 - m1, j2 + m2));
    if (vmax < vmin) return 0.0;
    double pref = sqrt((2.0*j3 + 1.0) * factd(j3+j1-j2) * factd(j3-j1+j2) *
                       factd(j1+j2-j3) / factd(j1+j2+j3+1));
    pref *= sqrt(factd(j3+m3)*factd(j3-m3)*factd(j1-m1)*factd(j1+m1)*
                 factd(j2-m2)*factd(j2+m2));
    double s = 0.0;
    for (int v = vmin; v <= vmax; ++v) {
        double t = 1.0 / (factd(v)*factd(j1+j2-j3-v)*factd(j1-m1-v)*
                          factd(j2+m2-v)*factd(j3-j2+m1+v)*factd(j3-j1-m2+v));
        s += (v & 1) ? -t : t;
    }
    return pref * s;
}

// q(l): complex change-of-basis (real <- complex SH), times (-i)^l.
static __device__ void build_q(int l, double2* q) {
    int n = 2*l + 1;
    for (int a = 0; a < n*n; ++a) { q[a].x = 0.0; q[a].y = 0.0; }
    const double s = 0.70710678118654752440;
    for (int m = -l; m < 0; ++m) {
        q[(l+m)*n + (l-m)].x = s;                    // column l+|m|
        q[(l+m)*n + (l+m)].y = -s;                   // column l-|m|, value -i*s
    }
    q[l*n + l].x = 1.0;
    for (int m = 1; m <= l; ++m) {
        double sgn = (m & 1) ? -1.0 : 1.0;
        q[(l+m)*n + (l+m)].x = sgn * s;
        q[(l+m)*n + (l-m)].y = sgn * s;              // +i * sgn * s
    }
    double fre, fim;                                  // (-i)^l
    switch (l & 3) {
        case 0: fre = 1;  fim = 0;  break;
        case 1: fre = 0;  fim = -1; break;
        case 2: fre = -1; fim = 0;  break;
        default: fre = 0; fim = 1;  break;
    }
    for (int a = 0; a < n*n; ++a) {
        double re = q[a].x, im = q[a].y;
        q[a].x = re*fre - im*fim;
        q[a].y = re*fim + im*fre;
    }
}

// One block per path: builds rows [PROFF,PROFF+2*l3+1) of the flattened
// cgW[99][16][16] table (r, i, j), zero outside the (l1,l2) block.
__global__ __launch_bounds__(128) void cg_init_kernel(float* __restrict__ cgW) {
    const int p  = blockIdx.x;
    const int l1 = PL1[p], l2 = PL2[p], l3 = PL3[p], rb = PROFF[p];
    const int n1 = 2*l1+1, n2 = 2*l2+1, n3 = 2*l3+1;
    const int t  = threadIdx.x;

    __shared__ double2 q1[49], q2[49], q3[49];
    __shared__ double  cg[245];
    __shared__ double2 D[245];
    __shared__ double  partR[128], partI[128];
    __shared__ double  nrm2[2];

    if (t == 0) build_q(l1, q1);
    if (t == 1) build_q(l2, q2);
    if (t == 2) build_q(l3, q3);
    const int tot = n1 * n2 * n3;
    for (int idx = t; idx < tot; idx += blockDim.x) {
        int k  = idx % n3;
        int jj = (idx / n3) % n2;
        int ii = idx / (n3 * n2);
        cg[idx] = cg_coeff(l1, ii - l1, l2, jj - l2, l3, k - l3);
    }
    __syncthreads();

    // D[i,j,k] = sum_{l,m,n} q1[i,l] q2[j,m] conj(q3[k,n]) cg[l,m,n]
    double locR = 0.0, locI = 0.0;
    for (int idx = t; idx < tot; idx += blockDim.x) {
        int k  = idx % n3;
        int jj = (idx / n3) % n2;
        int ii = idx / (n3 * n2);
        double dr = 0.0, di = 0.0;
        for (int l = 0; l < n1; ++l) {
            double ar = q1[ii*n1 + l].x, ai = q1[ii*n1 + l].y;
            for (int m = 0; m < n2; ++m) {
                double br = q2[jj*n2 + m].x, bi = q2[jj*n2 + m].y;
                double abr = ar*br - ai*bi;
                double abi = ar*bi + ai*br;
                for (int nn = 0; nn < n3; ++nn) {
                    double c = cg[(l*n2 + m)*n3 + nn];
                    if (c == 0.0) continue;
                    double qr = q3[k*n3 + nn].x, qi = -q3[k*n3 + nn].y;  // conj
                    dr += (abr*qr - abi*qi) * c;
                    di += (abr*qi + abi*qr) * c;
                }
            }
        }
        D[idx].x = dr; D[idx].y = di;
        locR += dr*dr; locI += di*di;
    }
    partR[t] = locR; partI[t] = locI;
    __syncthreads();
    if (t == 0) {
        double a = 0.0, b = 0.0;
        for (int i = 0; i < 128; ++i) { a += partR[i]; b += partI[i]; }
        nrm2[0] = a; nrm2[1] = b;
    }
    __syncthreads();

    const bool useRe = (nrm2[0] >= nrm2[1]);
    const double inv = 1.0 / sqrt(useRe ? nrm2[0] : nrm2[1]);
    const int io = SPHOFF[l1], jo = SPHOFF[l2];
    for (int idx = t; idx < n3 * 256; idx += blockDim.x) {
        int k  = idx >> 8;
        int i  = (idx >> 4) & 15;
        int j  = idx & 15;
        float val = 0.0f;
        int il = i - io, jl = j - jo;
        if (il >= 0 && il < n1 && jl >= 0 && jl < n2) {
            double2 d = D[(il*n2 + jl)*n3 + k];
            val = (float)((useRe ? d.x : d.y) * inv);
        }
        cgW[((rb + k)*16 + i)*16 + j] = val;
    }
}

// ------------------------------ main kernel --------------------------------
// Block = 256 threads = 8 waves = 2 edges. Per edge: 4 waves, each owning a
// 16-channel N-tile; 7 M-tiles x 4 K-steps of v_wmma_f32_16x16x4_f32.
__global__ __launch_bounds__(256) void tp_wmma_kernel(
        const float* __restrict__ x, const float* __restrict__ y,
        const float* __restrict__ cgW, float* __restrict__ out, int nE)
{
    __shared__ float Wsh[2][N_ROWS_PAD][W_PITCH];
    __shared__ float ysh[2][16];

    const int tid = threadIdx.x;
    const int e0  = blockIdx.x * 2;

    if (tid < 32) {
        int ee = tid >> 4, j = tid & 15;
        int e = e0 + ee;
        ysh[ee][j] = (e < nE) ? y[(size_t)e * 16 + j] : 0.0f;
    }
    __syncthreads();

    // Fold y into CG: W[r,i] = sum_j cgW[r,i,j] * y[j]; rows 99..111 padded 0.
    for (int idx = tid; idx < N_ROWS_PAD * 16; idx += 256) {
        int r = idx >> 4, i = idx & 15;
        float4 c0, c1, c2, c3;
        if (r < N_ROWS) {
            const float4* cg4 = (const float4*)(cgW + ((size_t)(r*16 + i)) * 16);
            c0 = cg4[0]; c1 = cg4[1]; c2 = cg4[2]; c3 = cg4[3];
        }
        #pragma unroll
        for (int ee = 0; ee < 2; ++ee) {
            float s = 0.0f;
            if (r < N_ROWS) {
                s  = c0.x*ysh[ee][0]  + c0.y*ysh[ee][1]  + c0.z*ysh[ee][2]  + c0.w*ysh[ee][3];
                s += c1.x*ysh[ee][4]  + c1.y*ysh[ee][5]  + c1.z*ysh[ee][6]  + c1.w*ysh[ee][7];
                s += c2.x*ysh[ee][8]  + c2.y*ysh[ee][9]  + c2.z*ysh[ee][10] + c2.w*ysh[ee][11];
                s += c3.x*ysh[ee][12] + c3.y*ysh[ee][13] + c3.z*ysh[ee][14] + c3.w*ysh[ee][15];
            }
            Wsh[ee][r][i] = s;
        }
    }
    __syncthreads();

    const int wave = tid >> 5;
    const int lane = tid & 31;
    const int ee   = wave >> 2;          // which of the 2 edges
    const int cb   = (wave & 3) * 16;    // channel tile base
    const int n    = lane & 15;          // N index within tile / M row for A
    const int kh   = lane >> 4;          // half-select: K offset 0 or 2
    const int e    = e0 + ee;
    if (e >= nE) return;                 // wave-uniform

    // B operand (x tile), K x N = 4 x 16 per step.
    // Layout: VGPR0 -> K = kb+0 (lanes 0-15) / kb+2 (lanes 16-31); VGPR1 -> +1.
    v2f B[4];
    const float* xe = x + (size_t)e * (N_EDGE_DIM * N_CH) + cb + n;
    #pragma unroll
    for (int ks = 0; ks < 4; ++ks) {
        B[ks].x = xe[(size_t)(ks*4 + 0 + 2*kh) * N_CH];
        B[ks].y = xe[(size_t)(ks*4 + 1 + 2*kh) * N_CH];
    }

    float* oe = out + (size_t)e * (N_ROWS * N_CH) + cb + n;
    #pragma unroll
    for (int mt = 0; mt < 7; ++mt) {
        v8f acc = {0.f, 0.f, 0.f, 0.f, 0.f, 0.f, 0.f, 0.f};
        #pragma unroll
        for (int ks = 0; ks < 4; ++ks) {
            // A operand (W tile), M x K = 16 x 4: lane -> row M = n,
            // VGPR0/1 -> K = kb + {0,1} (lanes 0-15) or kb + {2,3} (16-31).
            v2f A;
            A.x = Wsh[ee][mt*16 + n][ks*4 + 0 + 2*kh];
            A.y = Wsh[ee][mt*16 + n][ks*4 + 1 + 2*kh];
            acc = __builtin_amdgcn_wmma_f32_16x16x4_f32(
                    /*neg_a=*/false, A, /*neg_b=*/false, B[ks],
                    /*c_mod=*/(short)0, acc, /*reuse_a=*/false, /*reuse_b=*/false);
        }
        // C/D layout: VGPR v -> M = v (lanes 0-15) / v+8 (lanes 16-31).
        #pragma unroll
        for (int v = 0; v < 8; ++v) {
            int r = mt*16 + v + 8*kh;
            if (r < N_ROWS) oe[(size_t)r * N_CH] = acc[v];
        }
    }
}

// ------------------------------- launcher ----------------------------------

extern "C" void kernel_launch(void* const* d_in, const int* in_sizes, int n_in,
                              void* d_out, int out_size, void* d_ws, size_t ws_size,
                              hipStream_t stream) {
    const float* x = (const float*)d_in[0];   // (E,16,64) f32
    const float* y = (const float*)d_in[1];   // (E,16)    f32
    float* out  = (float*)d_out;              // (E,99,64) f32
    float* cgW  = (float*)d_ws;               // 99*16*16 f32 table

    const int E = in_sizes[1] / 16;

    cg_init_kernel<<<N_PATHS, 128, 0, stream>>>(cgW);
    tp_wmma_kernel<<<dim3((E + 1) / 2), 256, 0, stream>>>(x, y, cgW, out, E);
}